// VNCMDLoss_249108103965
// MI455X (gfx1250) — compile-verified
//
#include <hip/hip_runtime.h>

// ---------------------------------------------------------------------------
// VNCMD loss: memory-bound multi-term reduction.
//   recon_loss = mean((rr-tr)^2) + mean((ri-ti)^2)                 over B*T
//   if_loss    = mean((mask*(eIF - tif))^2)                        over B*N*T
//   smooth     = sum_bn mask * mean_t((eIF[t+1]-eIF[t])^2) / max(count,1)
//   total      = recon + 0.5*if + (count>0 ? 0.1*smooth : 0)
// Reductions use V_WMMA_F32_16X16X4_F32 for the 32-lane cross-lane fold.
// ---------------------------------------------------------------------------

typedef __attribute__((ext_vector_type(2))) float v2f;
typedef __attribute__((ext_vector_type(8))) float v8f;

constexpr int   kB   = 8;
constexpr int   kN   = 8;
constexpr int   kT   = 262144;          // power of two
constexpr int   kVecPerRow = kT / 4;    // 65536 = 1<<16
constexpr long long kBT  = (long long)kB * kT;            // 2,097,152
constexpr long long kBNT = (long long)kB * kN * kT;       // 16,777,216

constexpr int BLK_RECON = 512;   // partial slots for kernel 1
constexpr int BLK_IF    = 1024;  // partial slots for kernel 2
constexpr int NTHREADS  = 256;   // 8 waves (wave32)

// ---------------------------------------------------------------------------
// Reduce two per-lane scalars across the full wave32 with ONE wmma.
// A matrix (16x4 f32): a.x -> K=0 (lanes 0-15) / K=2 (lanes 16-31)
//                      a.y -> K=1 (lanes 0-15) / K=3 (lanes 16-31)
// B matrix (4x16 f32): selector: columns 0-7 pick s1 (K0,K2), cols 8-15 pick
// s2 (K1,K3). So D[m][n<8] = s1[m]+s1[m+16], D[m][n>=8] = s2[m]+s2[m+16].
// Summing the 8 C/D VGPRs per lane folds over m; two lane reads finish it.
// ---------------------------------------------------------------------------
__device__ __forceinline__ void wave_reduce2(float s1, float s2,
                                             float& r1, float& r2) {
  const int lane = threadIdx.x & 31;
  v2f a;
  a.x = s1;
  a.y = s2;
  const float sel = ((lane & 15) < 8) ? 1.0f : 0.0f;
  v2f b;
  b.x = sel;          // B rows K=0,K=2: select s1 for columns 0..7
  b.y = 1.0f - sel;   // B rows K=1,K=3: select s2 for columns 8..15
  v8f c = {};
  c = __builtin_amdgcn_wmma_f32_16x16x4_f32(
      /*neg_a=*/false, a, /*neg_b=*/false, b,
      /*c_mod=*/(short)0, c, /*reuse_a=*/false, /*reuse_b=*/false);
  // fold over M (rows): lanes 0-15 hold M=r, lanes 16-31 hold M=r+8
  float acc = c[0] + c[1] + c[2] + c[3] + c[4] + c[5] + c[6] + c[7];
  // acc[lane<8]   = sum_{m=0..7} s1[m]+s1[m+16];  acc[8..15]  same for s2
  // acc[16..23]   = sum_{m=8..15} s1 part;        acc[24..31] same for s2
  r1 = __shfl(acc, 0, 32) + __shfl(acc, 16, 32);
  r2 = __shfl(acc, 8, 32) + __shfl(acc, 24, 32);
}

// Block-level (256 threads = 8 waves) reduction of two scalars.
// Result valid on threadIdx.x == 0 only.
__device__ __forceinline__ void block_reduce2(float s1, float s2,
                                              float& t1, float& t2) {
  __shared__ float l1[NTHREADS / 32];
  __shared__ float l2[NTHREADS / 32];
  __syncthreads();  // protect LDS reuse across repeated calls
  float r1, r2;
  wave_reduce2(s1, s2, r1, r2);
  const int wid  = threadIdx.x >> 5;
  const int lane = threadIdx.x & 31;
  if (lane == 0) { l1[wid] = r1; l2[wid] = r2; }
  __syncthreads();
  t1 = 0.0f;
  t2 = 0.0f;
  if (threadIdx.x == 0) {
#pragma unroll
    for (int i = 0; i < NTHREADS / 32; ++i) { t1 += l1[i]; t2 += l2[i]; }
  }
}

// ---------------------------------------------------------------------------
// Kernel 1: sum (recon_real-target_real)^2 + (recon_imag-target_imag)^2
// ---------------------------------------------------------------------------
__global__ __launch_bounds__(NTHREADS)
void vncmd_recon_partial(const float4* __restrict__ rr,
                         const float4* __restrict__ ri,
                         const float4* __restrict__ tr,
                         const float4* __restrict__ ti,
                         float* __restrict__ part, int nvec) {
  const int stride = gridDim.x * blockDim.x;
  float s = 0.0f;
  for (int v = blockIdx.x * blockDim.x + threadIdx.x; v < nvec; v += stride) {
    float4 a = rr[v];
    float4 b = tr[v];
    float dx = a.x - b.x, dy = a.y - b.y, dz = a.z - b.z, dw = a.w - b.w;
    s += dx * dx + dy * dy + dz * dz + dw * dw;
    float4 c = ri[v];
    float4 d = ti[v];
    dx = c.x - d.x; dy = c.y - d.y; dz = c.z - d.z; dw = c.w - d.w;
    s += dx * dx + dy * dy + dz * dz + dw * dw;
  }
  float t1, t2;
  block_reduce2(s, 0.0f, t1, t2);
  if (threadIdx.x == 0) part[blockIdx.x] = t1;
}

// ---------------------------------------------------------------------------
// Kernel 2: masked IF mse + masked smoothness sums over (B*N, T).
// Rows with mask==0 contribute nothing -> skip their loads entirely
// (wave-uniform branch: a wave's 32 consecutive float4s sit in one row
// except at the 1MB row boundary).
// ---------------------------------------------------------------------------
__global__ __launch_bounds__(NTHREADS)
void vncmd_if_partial(const float* __restrict__ eIF,
                      const float* __restrict__ tif,
                      const int* __restrict__ mask,
                      float* __restrict__ partIf,
                      float* __restrict__ partSm, int nvec) {
  const float4* __restrict__ e4 = (const float4*)eIF;
  const float4* __restrict__ t4 = (const float4*)tif;
  const int stride = gridDim.x * blockDim.x;
  float sIf = 0.0f;
  float sSm = 0.0f;
  for (int v = blockIdx.x * blockDim.x + threadIdx.x; v < nvec; v += stride) {
    const int row = v >> 16;            // v / (T/4)
    const int vr  = v & (kVecPerRow - 1);
    if (mask[row] == 1) {
      float4 e = e4[v];
      float4 t = t4[v];
      float d0 = e.x - t.x, d1 = e.y - t.y, d2 = e.z - t.z, d3 = e.w - t.w;
      sIf += d0 * d0 + d1 * d1 + d2 * d2 + d3 * d3;
      float f0 = e.y - e.x, f1 = e.z - e.y, f2 = e.w - e.z;
      float sm = f0 * f0 + f1 * f1 + f2 * f2;
      if (vr + 1 < kVecPerRow) {        // cross-vector diff within the row
        float enext = eIF[(size_t)v * 4 + 4];
        float f3 = enext - e.w;
        sm += f3 * f3;
      }
      sSm += sm;
    }
  }
  float t1, t2;
  block_reduce2(sIf, sSm, t1, t2);
  if (threadIdx.x == 0) {
    partIf[blockIdx.x] = t1;
    partSm[blockIdx.x] = t2;
  }
}

// ---------------------------------------------------------------------------
// Kernel 3: fold block partials + mask count, emit the 4 scalars.
// ---------------------------------------------------------------------------
__global__ __launch_bounds__(NTHREADS)
void vncmd_finalize(const float* __restrict__ partR, int nR,
                    const float* __restrict__ partIf,
                    const float* __restrict__ partSm, int nB,
                    const int* __restrict__ mask,
                    float* __restrict__ out) {
  float sR = 0.0f, sI = 0.0f, sS = 0.0f, cnt = 0.0f;
  for (int i = threadIdx.x; i < nR; i += blockDim.x) sR += partR[i];
  for (int i = threadIdx.x; i < nB; i += blockDim.x) {
    sI += partIf[i];
    sS += partSm[i];
  }
  for (int i = threadIdx.x; i < kB * kN; i += blockDim.x)
    cnt += (mask[i] == 1) ? 1.0f : 0.0f;

  float tR, tI, tS, tC;
  block_reduce2(sR, sI, tR, tI);
  block_reduce2(sS, cnt, tS, tC);

  if (threadIdx.x == 0) {
    float recon_loss  = tR / (float)kBT;
    float if_loss     = tI / (float)kBNT;
    float denom       = (tC > 1.0f) ? tC : 1.0f;
    float smooth_loss = tS / ((float)(kT - 1) * denom);
    float total = 1.0f * recon_loss + 0.5f * if_loss +
                  ((tC > 0.0f) ? 0.1f * smooth_loss : 0.0f);
    out[0] = total;
    out[1] = recon_loss;
    out[2] = if_loss;
    out[3] = smooth_loss;
  }
}

// ---------------------------------------------------------------------------
extern "C" void kernel_launch(void* const* d_in, const int* in_sizes, int n_in,
                              void* d_out, int out_size, void* d_ws,
                              size_t ws_size, hipStream_t stream) {
  const float* rr   = (const float*)d_in[0];  // recon_real  (B,T)
  const float* ri   = (const float*)d_in[1];  // recon_imag  (B,T)
  const float* tr   = (const float*)d_in[2];  // target_real (B,T)
  const float* ti   = (const float*)d_in[3];  // target_imag (B,T)
  const float* eIF  = (const float*)d_in[4];  // eIF       (B,N,T)
  const float* tif  = (const float*)d_in[5];  // target_if (B,N,T)
  const int*   mask = (const int*)d_in[6];    // mode_mask (B,N)

  float* ws     = (float*)d_ws;
  float* partR  = ws;                        // BLK_RECON floats
  float* partIf = ws + BLK_RECON;            // BLK_IF floats
  float* partSm = ws + BLK_RECON + BLK_IF;   // BLK_IF floats

  const int nvecR  = (int)(kBT / 4);   // 524288
  const int nvecIF = (int)(kBNT / 4);  // 4194304

  vncmd_recon_partial<<<BLK_RECON, NTHREADS, 0, stream>>>(
      (const float4*)rr, (const float4*)ri, (const float4*)tr,
      (const float4*)ti, partR, nvecR);

  vncmd_if_partial<<<BLK_IF, NTHREADS, 0, stream>>>(
      eIF, tif, mask, partIf, partSm, nvecIF);

  vncmd_finalize<<<1, NTHREADS, 0, stream>>>(
      partR, BLK_RECON, partIf, partSm, BLK_IF, mask, (float*)d_out);
}